// GridSampler_51007031607828
// MI455X (gfx1250) — compile-verified
//
#include <hip/hip_runtime.h>

typedef __attribute__((ext_vector_type(2))) float v2f;
typedef __attribute__((ext_vector_type(4))) float v4f;
typedef __attribute__((ext_vector_type(8))) float v8f;

#define NB  2
#define NR  4096
#define NP  64
#define ND  16
#define NH  256
#define NW  256
#define NCF 64
#define DHW (ND * NH * NW)   // 1,048,576 elements per channel

// ---------------------------------------------------------------------------
// Kernel 1: per-ray affine transform via V_WMMA_F32_16X16X4_F32.
// o_loc = [ox oy oz 1] * M (4x4, rows 0..2 = rotation, row 3 = translation)
// d_loc = [dx dy dz 0] * M
// One wave = 16 rays. A layout (32-bit 16x4): lanes 0-15 hold K=0,1 in
// VGPR0,1; lanes 16-31 hold K=2,3.  B (4x16): lanes = N, K split the same way.
// D (16x16 f32): VGPR r -> row r (lanes 0-15) / row r+8 (lanes 16-31).
// ---------------------------------------------------------------------------
__global__ __launch_bounds__(32) void ray_transform_wmma(
    const float* __restrict__ w2l,      // [B,4,4]
    const float* __restrict__ origins,  // [B*R,3]
    const float* __restrict__ dirs,     // [B*R,3]
    float* __restrict__ oloc,           // [B*R,3]
    float* __restrict__ dloc)           // [B*R,3]
{
  const int lane = threadIdx.x;        // 0..31
  const int base = blockIdx.x * 16;    // first ray of this wave's tile
  const int m    = lane & 15;
  const int rayM = base + m;
  const int b    = base / NR;          // tile never straddles a batch (R%16==0)

  // A operands (homogeneous ray rows)
  const float* o3 = origins + (size_t)rayM * 3;
  const float* d3 = dirs    + (size_t)rayM * 3;
  v2f Ao, Ad;
  if (lane < 16) { Ao.x = o3[0]; Ao.y = o3[1]; Ad.x = d3[0]; Ad.y = d3[1]; }
  else           { Ao.x = o3[2]; Ao.y = 1.0f;  Ad.x = d3[2]; Ad.y = 0.0f; }

  // B operand: B[k][n] = M[k][n] for n<3, else 0
  v2f Bm; Bm.x = 0.0f; Bm.y = 0.0f;
  const float* M = w2l + b * 16;
  if (m < 3) {
    if (lane < 16) { Bm.x = M[0 * 4 + m]; Bm.y = M[1 * 4 + m]; }
    else           { Bm.x = M[2 * 4 + m]; Bm.y = M[3 * 4 + m]; }
  }

  v8f Z = {};
  v8f Do = __builtin_amdgcn_wmma_f32_16x16x4_f32(false, Ao, false, Bm,
                                                 (short)0, Z, false, false);
  v8f Dd = __builtin_amdgcn_wmma_f32_16x16x4_f32(false, Ad, false, Bm,
                                                 (short)0, Z, false, false);

  if (m < 3) {
    const int rayOut0 = base + ((lane < 16) ? 0 : 8);
#pragma unroll
    for (int r = 0; r < 8; ++r) {
      oloc[(size_t)(rayOut0 + r) * 3 + m] = Do[r];
      dloc[(size_t)(rayOut0 + r) * 3 + m] = Dd[r];
    }
  }
}

// ---------------------------------------------------------------------------
// Kernel 2: fused trilinear gather. One lane = one sample point.
// Corner offsets + validity-masked weights computed once, reused for all 65
// channels. Feature output is 64 contiguous floats per lane -> 16x b128 NT
// stores (keep the 192MB L2 for the 520MB volume, not for write-once output).
// ---------------------------------------------------------------------------
__global__ __launch_bounds__(256) void grid_sample_kernel(
    const float* __restrict__ density,   // [B,1,D,H,W]
    const float* __restrict__ features,  // [B,64,D,H,W]
    const float* __restrict__ oloc,      // [B*R,3]
    const float* __restrict__ dloc,      // [B*R,3]
    const float* __restrict__ lengths,   // [B,R,P]
    float* __restrict__ outD,            // [B*R*P]
    float* __restrict__ outF)            // [B*R*P,64]
{
  const int g = blockIdx.x * blockDim.x + threadIdx.x;   // point id
  if (g >= NB * NR * NP) return;
  const int ray = g >> 6;     // NP == 64
  const int b   = ray >> 12;  // NR == 4096

  const float t = lengths[g];
  const float* ov = oloc + (size_t)ray * 3;
  const float* dv = dloc + (size_t)ray * 3;
  const float x = ov[0] + dv[0] * t;
  const float y = ov[1] + dv[1] * t;
  const float z = ov[2] + dv[2] * t;

  // normalized [-1,1] -> voxel coords (align_corners=True)
  const float fx = (x + 1.0f) * 0.5f * (float)(NW - 1);
  const float fy = (y + 1.0f) * 0.5f * (float)(NH - 1);
  const float fz = (z + 1.0f) * 0.5f * (float)(ND - 1);
  const float flx = floorf(fx), fly = floorf(fy), flz = floorf(fz);
  const float tx = fx - flx, ty = fy - fly, tz = fz - flz;
  const int ix = (int)flx, iy = (int)fly, iz = (int)flz;
  const int ix1 = ix + 1, iy1 = iy + 1, iz1 = iz + 1;

  // validity-masked per-axis weights + clamped indices (padding_mode='zeros')
  const float wx0 = (1.0f - tx) * ((ix  >= 0 && ix  < NW) ? 1.0f : 0.0f);
  const float wx1 = tx          * ((ix1 >= 0 && ix1 < NW) ? 1.0f : 0.0f);
  const float wy0 = (1.0f - ty) * ((iy  >= 0 && iy  < NH) ? 1.0f : 0.0f);
  const float wy1 = ty          * ((iy1 >= 0 && iy1 < NH) ? 1.0f : 0.0f);
  const float wz0 = (1.0f - tz) * ((iz  >= 0 && iz  < ND) ? 1.0f : 0.0f);
  const float wz1 = tz          * ((iz1 >= 0 && iz1 < ND) ? 1.0f : 0.0f);
  const int cx0 = min(max(ix,  0), NW - 1), cx1 = min(max(ix1, 0), NW - 1);
  const int cy0 = min(max(iy,  0), NH - 1), cy1 = min(max(iy1, 0), NH - 1);
  const int cz0 = min(max(iz,  0), ND - 1), cz1 = min(max(iz1, 0), ND - 1);

  int   offs[8];
  float wts[8];
  const int r00 = (cz0 * NH + cy0) * NW, r01 = (cz0 * NH + cy1) * NW;
  const int r10 = (cz1 * NH + cy0) * NW, r11 = (cz1 * NH + cy1) * NW;
  offs[0] = r00 + cx0;  wts[0] = wz0 * wy0 * wx0;
  offs[1] = r00 + cx1;  wts[1] = wz0 * wy0 * wx1;
  offs[2] = r01 + cx0;  wts[2] = wz0 * wy1 * wx0;
  offs[3] = r01 + cx1;  wts[3] = wz0 * wy1 * wx1;
  offs[4] = r10 + cx0;  wts[4] = wz1 * wy0 * wx0;
  offs[5] = r10 + cx1;  wts[5] = wz1 * wy0 * wx1;
  offs[6] = r11 + cx0;  wts[6] = wz1 * wy1 * wx0;
  offs[7] = r11 + cx1;  wts[7] = wz1 * wy1 * wx1;

  // density (1 channel)
  {
    const float* dp = density + (size_t)b * DHW;
    float acc = 0.0f;
#pragma unroll
    for (int k = 0; k < 8; ++k) acc = fmaf(wts[k], dp[offs[k]], acc);
    __builtin_nontemporal_store(acc, outD + g);
  }

  // 64 feature channels, 4 at a time -> b128 non-temporal stores
  const float* fb = features + (size_t)b * NCF * DHW;
  v4f* of4 = (v4f*)(outF + (size_t)g * NCF);
#pragma unroll
  for (int cc = 0; cc < NCF; cc += 4) {
    v4f v;
#pragma unroll
    for (int j = 0; j < 4; ++j) {
      const float* fp = fb + (size_t)(cc + j) * DHW;
      float acc = 0.0f;
#pragma unroll
      for (int k = 0; k < 8; ++k) acc = fmaf(wts[k], fp[offs[k]], acc);
      v[j] = acc;
    }
    __builtin_nontemporal_store(v, of4 + (cc >> 2));
  }
}

// ---------------------------------------------------------------------------
extern "C" void kernel_launch(void* const* d_in, const int* in_sizes, int n_in,
                              void* d_out, int out_size, void* d_ws, size_t ws_size,
                              hipStream_t stream) {
  const float* density  = (const float*)d_in[0];
  const float* features = (const float*)d_in[1];
  const float* w2l      = (const float*)d_in[2];
  const float* origins  = (const float*)d_in[3];
  const float* dirs     = (const float*)d_in[4];
  const float* lengths  = (const float*)d_in[5];
  float* out  = (float*)d_out;

  float* oloc = (float*)d_ws;                       // [B*R,3]
  float* dloc = oloc + (size_t)NB * NR * 3;         // [B*R,3]  (192KB total)

  // Stage 1: 512 full waves, 16 rays each, WMMA transform.
  ray_transform_wmma<<<(NB * NR) / 16, 32, 0, stream>>>(w2l, origins, dirs,
                                                        oloc, dloc);

  // Stage 2: fused trilinear gather (memory-bound main kernel).
  const int Ntot = NB * NR * NP;                    // 524,288 points
  grid_sample_kernel<<<Ntot / 256, 256, 0, stream>>>(
      density, features, oloc, dloc, lengths, out, out + Ntot);
}